// DCTEmbedder_88794153877654
// MI455X (gfx1250) — compile-verified
//
#include <hip/hip_runtime.h>
#include <hip/hip_bf16.h>

// ---------------------------------------------------------------------------
// DCT steganography embedder for MI455X (gfx1250), wave32 + WMMA f32 path.
//
// Shapes: cover [8,3,512,512] f32, secret_bits [8,4096] i32
// Outputs (concat): stego [8,3,512,512] f32, embedding_map [8,3,64,64,8,8] f32
// ---------------------------------------------------------------------------

typedef __attribute__((ext_vector_type(2))) float v2f;
typedef __attribute__((ext_vector_type(8))) float v8f;

#define NBLK   98304      // 8*3*64*64 blocks of 8x8
#define NTILE  24576      // 8*3*32*32 tiles of 16x16 (2x2 blocks each)
#define KQ     29490      // floor(0.3 * (NBLK-1)); frac = 0.9
#define IMGSZ  (512*512)
#define NPIX   6291456    // 8*3*512*512

// 8x8 orthonormal DCT-II matrix as literals: D[i][j], i=frequency, j=sample.
// Row 0 = sqrt(1/8); rows 1..7 = 0.5*cos((2j+1)*i*pi/16), h_n = 0.5*cos(n*pi/16).
#define H0 0.35355339059327373f
#define H1 0.49039264020161522f
#define H2 0.46193976625564337f
#define H3 0.41573480615127262f
#define H5 0.27778511650980114f
#define H6 0.19134171618254492f
#define H7 0.09754516100806417f
__device__ __constant__ float D8TAB[64] = {
    H0,  H0,  H0,  H0,  H0,  H0,  H0,  H0,
    H1,  H3,  H5,  H7, -H7, -H5, -H3, -H1,
    H2,  H6, -H6, -H2, -H2, -H6,  H6,  H2,
    H3, -H7, -H1, -H5,  H5,  H1,  H7, -H3,
    H0, -H0, -H0,  H0,  H0, -H0, -H0,  H0,
    H5, -H1,  H7,  H3, -H3, -H7,  H1, -H5,
    H6, -H2,  H2, -H6, -H6,  H2, -H2,  H6,
    H7, -H5,  H3, -H1,  H1, -H3,  H5, -H7,
};

// blockdiag(D, D) 16x16
__device__ __forceinline__ float dblk(int m, int k) {
    return ((m >> 3) == (k >> 3)) ? D8TAB[(m & 7) * 8 + (k & 7)] : 0.0f;
}

// D = A(16x4) * B(4x16) + C, f32 WMMA, chained 4x for K=16.
#define WMMA_F32(A, B, C) \
    __builtin_amdgcn_wmma_f32_16x16x4_f32(false, (A), false, (B), (short)0, (C), false, false)

// ---------------------------------------------------------------------------
// Kernel 1: per-8x8-block pixel variance.  One thread per block.
// ---------------------------------------------------------------------------
__global__ void var_kernel(const float* __restrict__ cover, float* __restrict__ var) {
    int i = blockIdx.x * blockDim.x + threadIdx.x;
    if (i >= NBLK) return;
    int bx = i & 63, by = (i >> 6) & 63, bc = i >> 12;      // bc = b*3 + c
    const float* base = cover + (size_t)bc * IMGSZ + (by * 8) * 512 + bx * 8;
    float s = 0.0f, ss = 0.0f;
    for (int r = 0; r < 8; ++r)
        for (int cc = 0; cc < 8; ++cc) {
            float x = base[r * 512 + cc];
            s += x; ss += x * x;
        }
    float mean = s * (1.0f / 64.0f);
    var[i] = ss * (1.0f / 64.0f) - mean * mean;
}

// ---------------------------------------------------------------------------
// Kernel 2: texture threshold = linear-interp quantile(var, 0.3).
// Order statistics via 32-step bisection on float-as-uint keys (var >= 0, so
// the uint bit pattern is order-preserving). Single workgroup of 1024.
// ---------------------------------------------------------------------------
__global__ void thr_kernel(const float* __restrict__ var, float* __restrict__ thrp) {
    __shared__ unsigned scnt[1024];
    const int tid = threadIdx.x;

    auto count_le = [&](unsigned limit) -> unsigned {
        unsigned c = 0;
        for (int i = tid; i < NBLK; i += 1024)
            if (__float_as_uint(var[i]) <= limit) ++c;
        scnt[tid] = c; __syncthreads();
        for (int off = 512; off > 0; off >>= 1) {
            if (tid < off) scnt[tid] += scnt[tid + off];
            __syncthreads();
        }
        unsigned total = scnt[0]; __syncthreads();
        return total;
    };
    auto select_rank = [&](unsigned r) -> float {   // r-th smallest (0-indexed)
        unsigned lo = 0u, hi = 0xFFFFFFFFu;
        while (lo < hi) {
            unsigned mid = lo + ((hi - lo) >> 1);
            if (count_le(mid) >= r + 1u) hi = mid; else lo = mid + 1u;
        }
        return __uint_as_float(lo);
    };

    float s0 = select_rank(KQ);
    float s1 = select_rank(KQ + 1);
    if (tid == 0) thrp[0] = s0 + 0.9f * (s1 - s0);
}

// ---------------------------------------------------------------------------
// Kernel 3: exclusive scan of texture flags over all 98304 blocks (each tex
// block contributes exactly 37 selected coefficients, so the 6.29M-element
// cumsum of sel collapses to this block-level scan). Single workgroup.
// ---------------------------------------------------------------------------
__global__ void scan_kernel(const float* __restrict__ var, const float* __restrict__ thrp,
                            int* __restrict__ cum) {
    __shared__ int sb[1024];
    const int tid = threadIdx.x;
    const float thr = thrp[0];
    int carry = 0;
    for (int chunk = 0; chunk < NBLK / 1024; ++chunk) {
        int i = chunk * 1024 + tid;
        int t = (var[i] > thr) ? 1 : 0;
        sb[tid] = t; __syncthreads();
        for (int off = 1; off < 1024; off <<= 1) {
            int v = (tid >= off) ? sb[tid - off] : 0; __syncthreads();
            sb[tid] += v; __syncthreads();
        }
        cum[i] = carry + sb[tid] - t;        // exclusive count of tex blocks before i
        int total = sb[1023]; __syncthreads();
        carry += total;
    }
}

// ---------------------------------------------------------------------------
// Kernel 4 (fused): forward DCT -> LSB embedding -> inverse DCT, one 16x16
// tile (= 2x2 DCT blocks) per wave, 16x v_wmma_f32_16x16x4_f32 per tile.
//
// Operand layouts per CDNA5 ISA 7.12.2 (f32, wave32), lo=lane%16, hi=lane/16:
//   A chunk k0:  a.x = A[lo][k0+2*hi],  a.y = A[lo][k0+2*hi+1]
//   B chunk k0:  b.x = B[k0+2*hi][lo],  b.y = B[k0+2*hi+1][lo]
//   C/D:         d[r] = D[r+8*hi][lo]
// LDS roundtrips convert D-layout results back into A/B operand layouts.
//
// Block-local coords: gy0,gx0 are multiples of 16 => u = r (compile-time per
// unrolled iteration), v = lane%8 (loop-invariant). Mid-band rank has closed
// form per row; blockId/var/cum lookups hoist out of the per-coeff loop.
// ---------------------------------------------------------------------------
__global__ __launch_bounds__(256) void dct_embed_idct_kernel(
    const float* __restrict__ cover, const int* __restrict__ bits,
    const float* __restrict__ var,   const int* __restrict__ cum,
    const float* __restrict__ thrp,
    float* __restrict__ stego, float* __restrict__ emap)
{
    __shared__ float lds[8][256];
    const int wave = threadIdx.x >> 5;
    const int lane = threadIdx.x & 31;
    const int lo = lane & 15, hi = lane >> 4;
    float* T = lds[wave];

    const int tile = blockIdx.x * 8 + wave;                 // 0..NTILE-1
    const int txT = tile & 31;
    const int tyT = (tile >> 5) & 31;
    const int bc  = tile >> 10;                             // b*3 + c
    const int b   = bc / 3;
    const size_t imgbase = (size_t)bc * IMGSZ;
    const int gx0 = txT * 16, gy0 = tyT * 16;
    const float thr = thrp[0];

    // Warm the cache lines for the later K-chunks during operand setup.
    __builtin_prefetch(&cover[imgbase + (size_t)(gy0 + 8 + 2 * hi) * 512 + gx0 + lo], 0, 3);
    __builtin_prefetch(&cover[imgbase + (size_t)(gy0 + 12 + 2 * hi) * 512 + gx0 + lo], 0, 3);

    // Constant operands: cA[j] = blockdiag(D,D) in A-layout (also Dblk^T in
    // B-layout); cAT[j] = Dblk^T in A-layout (also Dblk in B-layout).
    v2f cA[4], cAT[4];
    for (int j = 0; j < 4; ++j) {
        int k0 = 4 * j;
        cA[j].x  = dblk(lo, k0 + 2 * hi);
        cA[j].y  = dblk(lo, k0 + 2 * hi + 1);
        cAT[j].x = dblk(k0 + 2 * hi, lo);
        cAT[j].y = dblk(k0 + 2 * hi + 1, lo);
    }

    // ---- Forward stage 1: Y = Dblk * X  (B operand streamed from global) --
    v8f y = {};
    for (int j = 0; j < 4; ++j) {
        int k0 = 4 * j;
        v2f bx;
        bx.x = cover[imgbase + (size_t)(gy0 + k0 + 2 * hi)     * 512 + gx0 + lo];
        bx.y = cover[imgbase + (size_t)(gy0 + k0 + 2 * hi + 1) * 512 + gx0 + lo];
        y = WMMA_F32(cA[j], bx, y);
    }
    // D-layout -> LDS
    for (int r = 0; r < 8; ++r) T[(r + 8 * hi) * 16 + lo] = y[r];
    // ---- Forward stage 2: Z = Y * Dblk^T  (A from LDS, B = cA) ------------
    v8f z = {};
    for (int j = 0; j < 4; ++j) {
        int k0 = 4 * j;
        v2f a;
        a.x = T[lo * 16 + k0 + 2 * hi];
        a.y = T[lo * 16 + k0 + 2 * hi + 1];
        z = WMMA_F32(a, cA[j], z);
    }

    // ---- Embedding: each lane owns 8 unique coefficients of the tile ------
    // Per-row (u) band constants for 3 <= u+v <= 8:
    //   vLo[u] = max(3-u,0), vHi[u] = min(8-u,7), rowBase[u] = cumulative count
    {
        const int vcol    = lo & 7;                         // v (loop-invariant)
        const int blockId = (bc * 64 + (tyT * 2 + hi)) * 64 + (txT * 2 + (lo >> 3));
        const bool tex    = var[blockId] > thr;
        const int  base   = 37 * cum[blockId];              // selected coeffs before block
        const int* bbits  = bits + b * 4096;
        float* emapBlk    = emap + (size_t)blockId * 64 + vcol;

        const int vLo[8]     = {3, 2, 1, 0, 0, 0, 0, 0};
        const int vHi[8]     = {7, 7, 6, 5, 4, 3, 2, 1};
        const int rowBase[8] = {0, 5, 11, 17, 23, 28, 32, 35};

#pragma unroll
        for (int r = 0; r < 8; ++r) {                       // u = r
            const bool fm = (vcol >= vLo[r]) && (vcol <= vHi[r]);
            int inrow = vcol - vLo[r];
            inrow = inrow < 0 ? 0 : inrow;                  // clamp (only matters if !fm)
            const int order = base + rowBase[r] + inrow;
            const bool elig = tex && fm && (order < 4096);
            const int bidx  = (order < 4096) ? order : 4095;
            const float bit = (float)bbits[bidx];
            float coeff = z[r];
            // lsb = fmod(|rint(x)|, 2): exact integer parity without libm —
            // a - 2*floor(a*0.5) is exact for all fp32 integers (a >= 2^24 are even).
            const float a    = fabsf(rintf(coeff));         // jnp.round == rint (ties-even)
            const float lsb  = fmaf(-2.0f, floorf(a * 0.5f), a);
            const float sign = (coeff >= 0.0f) ? 1.0f : -1.0f;
            if (elig && (lsb != bit)) coeff += sign * (2.0f * bit - 1.0f) * 0.5f;
            z[r] = coeff;
            emapBlk[r * 8] = elig ? 1.0f : 0.0f;
        }
    }

    // ---- Inverse stage 1: W = Z * Dblk  (A = Z from LDS, B = cAT) ---------
    for (int r = 0; r < 8; ++r) T[(r + 8 * hi) * 16 + lo] = z[r];
    v8f w = {};
    for (int j = 0; j < 4; ++j) {
        int k0 = 4 * j;
        v2f a;
        a.x = T[lo * 16 + k0 + 2 * hi];
        a.y = T[lo * 16 + k0 + 2 * hi + 1];
        w = WMMA_F32(a, cAT[j], w);
    }
    // ---- Inverse stage 2: S = Dblk^T * W  (A = cAT, B = W from LDS) -------
    for (int r = 0; r < 8; ++r) T[(r + 8 * hi) * 16 + lo] = w[r];
    v8f s2 = {};
    for (int j = 0; j < 4; ++j) {
        int k0 = 4 * j;
        v2f bb;
        bb.x = T[(k0 + 2 * hi)     * 16 + lo];
        bb.y = T[(k0 + 2 * hi + 1) * 16 + lo];
        s2 = WMMA_F32(cAT[j], bb, s2);
    }
    for (int r = 0; r < 8; ++r)
        stego[imgbase + (size_t)(gy0 + r + 8 * hi) * 512 + gx0 + lo] = s2[r];
}

// ---------------------------------------------------------------------------
extern "C" void kernel_launch(void* const* d_in, const int* in_sizes, int n_in,
                              void* d_out, int out_size, void* d_ws, size_t ws_size,
                              hipStream_t stream) {
    (void)in_sizes; (void)n_in; (void)out_size; (void)ws_size;
    const float* cover = (const float*)d_in[0];
    const int*   bits  = (const int*)d_in[1];
    float* stego = (float*)d_out;
    float* emap  = (float*)d_out + NPIX;

    float* var  = (float*)d_ws;                                  // 98304 f32
    int*   cum  = (int*)((char*)d_ws + (size_t)NBLK * 4);        // 98304 i32
    float* thrp = (float*)((char*)d_ws + (size_t)NBLK * 8);      // 1 f32

    var_kernel<<<(NBLK + 255) / 256, 256, 0, stream>>>(cover, var);
    thr_kernel<<<1, 1024, 0, stream>>>(var, thrp);
    scan_kernel<<<1, 1024, 0, stream>>>(var, thrp, cum);
    dct_embed_idct_kernel<<<NTILE / 8, 256, 0, stream>>>(cover, bits, var, cum, thrp,
                                                         stego, emap);
}